// EMRouting2d_29540785061958
// MI455X (gfx1250) — compile-verified
//
#include <hip/hip_runtime.h>
#include <cstdint>

// EM routing 2d for MI455X (gfx1250), compile-only tuning.
// B=8, G=32, I=8, C=3, O=16, H=W=64, ITERS=3, f32 throughout.
// Block = 4 waves cooperating on one 16-pixel tile (shared 124 KB LDS).
#define HW_   4096
#define EPSF  1e-7f
#define PSTR  1636   // per-pixel stride in con_lds (dwords); 8*PSTR % 64 == 32 -> no half-wave conflicts
#define GST   17     // per-(c,g) stride; 16*GST % 64 == 16 -> halves hit disjoint banks

typedef float    v2f  __attribute__((ext_vector_type(2)));
typedef float    v8f  __attribute__((ext_vector_type(8)));
typedef uint32_t u32x4 __attribute__((ext_vector_type(4)));
typedef int      i32x4 __attribute__((ext_vector_type(4)));
typedef int      i32x8 __attribute__((ext_vector_type(8)));

__global__ __launch_bounds__(128) void em_routing_kernel(
    const float* __restrict__ x, const float* __restrict__ weight,
    const float* __restrict__ bias, float* __restrict__ out) {
  __shared__ float x_lds[256 * 16];     // 16 KB: x tile, [channel][pixel]
  __shared__ float con_lds[16 * PSTR];  // ~102 KB: con[pix][c][g][o] (padded)
  __shared__ float outbuf[48 * 16];     // 3 KB: staged output for coalesced store

  const int lane = threadIdx.x & 31;  // lane within wave
  const int w    = threadIdx.x >> 5;  // wave id, 0..3
  const int h    = lane >> 4;         // half-wave id
  const int o    = lane & 15;         // output channel within 16-tile / pixel row for A

  const int tile = blockIdx.x;        // one 16-pixel tile per block
  const int b    = tile >> 8;         // 256 tiles per image (4096/16)
  const int hw0  = (tile & 255) << 4;

  // ---- TDM: each wave async-loads its 64-channel x 16-pixel slice (64 rows,
  //      64 B/row, row stride 16 KB) into its x_lds slice. 4 TDM ops in flight.
  {
    uint64_t ga = (uint64_t)(uintptr_t)(x + (size_t)b * 256 * HW_ + (size_t)w * 64 * HW_ + hw0);
    uint32_t lds_off = (uint32_t)(uintptr_t)(&x_lds[w * 64 * 16]);  // low 32 bits == LDS offset
    u32x4 g0;
    g0[0] = 1u;                                   // count=1 (valid), user mode, no gather
    g0[1] = lds_off;                              // lds_addr [63:32]
    g0[2] = (uint32_t)ga;                         // global_addr [95:64]
    g0[3] = (uint32_t)((ga >> 32) & 0x01FFFFFFull) | (2u << 30);  // addr[56:32] | type=2
    i32x8 g1;
    g1[0] = 0x20000;               // data_size=2 (4 bytes), no multicast/pad/iterate
    g1[1] = (int)(4096u << 16);    // tensor_dim0[15:0] at bits 63:48
    g1[2] = (int)(64u << 16);      // tensor_dim1 = 64 rows in this slice
    g1[3] = (int)(16u << 16);      // tile_dim0 = 16 at bits 127:112
    g1[4] = 64;                    // tile_dim1 = 64 (tile_dim2 = 0)
    g1[5] = 4096;                  // tensor_dim0_stride[31:0]
    g1[6] = 0;                     // dim0_stride hi, dim1_stride lo (unused, 2D)
    g1[7] = 0;
    i32x4 gz = {0, 0, 0, 0};
#if __clang_major__ >= 23
    i32x8 gz8 = {0, 0, 0, 0, 0, 0, 0, 0};
    __builtin_amdgcn_tensor_load_to_lds(g0, g1, gz, gz, gz8, 0);
#else
    __builtin_amdgcn_tensor_load_to_lds(g0, g1, gz, gz, 0);
#endif
    __builtin_amdgcn_s_wait_tensorcnt(0);
    asm volatile("" ::: "memory");
  }
  __syncthreads();  // 4-wave WG: real s_barrier_signal/s_barrier_wait

  // ---- Phase 1: grouped 1x1 conv via V_WMMA_F32_16X16X4_F32; wave w -> groups [8w, 8w+8).
  // A(16x4): lane holds row M=o, K = koff..koff+1 in v0,v1 (ISA 7.12.2).
  // B(4x16): lane holds col N=o, K = koff..koff+1 in v0,v1 (mirrors A split).
  const int koff = h << 1;  // 0 or 2
  for (int g = w * 8; g < w * 8 + 8; ++g) {
    const int chb = g * 8 + koff;
    v2f a0, a1;
    a0.x = x_lds[(chb + 0) * 16 + o];
    a0.y = x_lds[(chb + 1) * 16 + o];
    a1.x = x_lds[(chb + 4) * 16 + o];
    a1.y = x_lds[(chb + 5) * 16 + o];
    v8f acc[3];
#pragma unroll
    for (int t = 0; t < 3; ++t) {
      const float* wp = weight + ((size_t)(t * 16 + o) * 256 + g * 8 + koff);
      v2f b0 = *(const v2f*)(wp);      // K = kk0: koff, koff+1   (global_load_b64, L2-hot)
      v2f b1 = *(const v2f*)(wp + 4);  // K = kk1
      v8f z = {0.f, 0.f, 0.f, 0.f, 0.f, 0.f, 0.f, 0.f};
      v8f d = __builtin_amdgcn_wmma_f32_16x16x4_f32(false, a0, false, b0,
                                                    (short)0, z, false, false);
      d = __builtin_amdgcn_wmma_f32_16x16x4_f32(false, a1, false, b1,
                                                (short)0, d, false, false);
      acc[t] = d;
    }
    // D layout: VGPR r, lane L: M = r + 8*(L>=16), N = L%16. Scatter to con_lds.
#pragma unroll
    for (int t = 0; t < 3; ++t)
#pragma unroll
      for (int rr = 0; rr < 8; ++rr)
        con_lds[(rr + 8 * h) * PSTR + (t * 32 + g) * GST + o] = acc[t][rr];
  }
  __syncthreads();

  // ---- Phase 2: EM routing, lane = (group-half h, o); wave w -> pixels [4w, 4w+4).
  const float bias0 = bias[o], bias1 = bias[16 + o], bias2 = bias[32 + o];

  for (int p = w * 4; p < w * 4 + 4; ++p) {
    float con[48], r[48];
#pragma unroll
    for (int c = 0; c < 3; ++c)
#pragma unroll
      for (int j = 0; j < 16; ++j)
        con[c * 16 + j] = con_lds[p * PSTR + (c * 32 + h * 16 + j) * GST + o];
#pragma unroll
    for (int k = 0; k < 48; ++k) r[k] = (1.0f / 3.0f);

    float mean[3];
    for (int it = 0; it < 3; ++it) {
      float rk[3], vn[3];
#pragma unroll
      for (int c = 0; c < 3; ++c) {
        float s0 = 0.f, s1 = 0.f, s2 = 0.f;
#pragma unroll
        for (int j = 0; j < 16; ++j) {  // in-lane sum over 16 groups
          float rv = r[c * 16 + j], cv = con[c * 16 + j];
          float rc = rv * cv;
          s0 += rv; s1 += rc; s2 += rc * cv;
        }
        s0 += __shfl_xor(s0, 16, 32);   // fold the other 16 groups
        s1 += __shfl_xor(s1, 16, 32);
        s2 += __shfl_xor(s2, 16, 32);
        rk[c]   = s0 + EPSF;
        mean[c] = s1 / rk[c];
        vn[c]   = s2;
      }
      if (it == 2) break;
      float var[3], lv[3], lpi[3];
#pragma unroll
      for (int c = 0; c < 3; ++c) {
        var[c] = vn[c] / rk[c] - mean[c] * mean[c] + EPSF;
        lv[c]  = __logf(var[c]);
        lpi[c] = __logf(rk[c] * (1.0f / 32.0f));  // log(pi_k)
      }
#pragma unroll
      for (int j = 0; j < 16; ++j) {  // softmax over C per (g,o)
        float t0, t1, t2, d;
        d = con[0 * 16 + j] - mean[0]; t0 = -0.5f * (d * d / var[0] + lv[0]) + lpi[0];
        d = con[1 * 16 + j] - mean[1]; t1 = -0.5f * (d * d / var[1] + lv[1]) + lpi[1];
        d = con[2 * 16 + j] - mean[2]; t2 = -0.5f * (d * d / var[2] + lv[2]) + lpi[2];
        float m  = fmaxf(t0, fmaxf(t1, t2));
        float e0 = __expf(t0 - m), e1 = __expf(t1 - m), e2 = __expf(t2 - m);
        float inv = 1.0f / (e0 + e1 + e2);
        r[0 * 16 + j] = e0 * inv;
        r[1 * 16 + j] = e1 * inv;
        r[2 * 16 + j] = e2 * inv;
      }
    }
    if (h == 0) {  // mean replicated in both halves; one writer per (o, p)
      outbuf[(0 * 16 + o) * 16 + p] = mean[0] + bias0;
      outbuf[(1 * 16 + o) * 16 + p] = mean[1] + bias1;
      outbuf[(2 * 16 + o) * 16 + p] = mean[2] + bias2;
    }
  }
  __syncthreads();

  // ---- coalesced store: 48 channels x 16 contiguous pixels, all 128 lanes ----
  for (int idx = threadIdx.x; idx < 768; idx += 128) {
    int j = idx >> 4, p = idx & 15;
    out[((size_t)b * 48 + j) * HW_ + hw0 + p] = outbuf[idx];
  }
}

extern "C" void kernel_launch(void* const* d_in, const int* in_sizes, int n_in,
                              void* d_out, int out_size, void* d_ws, size_t ws_size,
                              hipStream_t stream) {
  const float* x    = (const float*)d_in[0];
  const float* w    = (const float*)d_in[1];
  const float* bias = (const float*)d_in[2];
  float* out        = (float*)d_out;
  const int Bn = in_sizes[0] / (256 * HW_);   // batch from x element count
  dim3 grid(Bn * (HW_ / 16));                 // one 16-pixel tile per 4-wave block
  em_routing_kernel<<<grid, dim3(128), 0, stream>>>(x, w, bias, out);
}